// ReprogrammingLayer_52527450030232
// MI455X (gfx1250) — compile-verified
//
#include <hip/hip_runtime.h>

typedef __attribute__((ext_vector_type(2))) float v2f;
typedef __attribute__((ext_vector_type(8))) float v8f;

static __device__ __forceinline__ v8f wmma_f32(v2f a, v2f b, v8f c) {
  // D = A(16x4) * B(4x16) + C(16x16), fp32, wave32
  return __builtin_amdgcn_wmma_f32_16x16x4_f32(false, a, false, b, (short)0, c,
                                               false, false);
}

static __device__ __forceinline__ float half_max(float x) {
  x = fmaxf(x, __shfl_xor(x, 1, 32));
  x = fmaxf(x, __shfl_xor(x, 2, 32));
  x = fmaxf(x, __shfl_xor(x, 4, 32));
  x = fmaxf(x, __shfl_xor(x, 8, 32));
  return x;  // max over each 16-lane half
}
static __device__ __forceinline__ float half_sum(float x) {
  x += __shfl_xor(x, 1, 32);
  x += __shfl_xor(x, 2, 32);
  x += __shfl_xor(x, 4, 32);
  x += __shfl_xor(x, 8, 32);
  return x;  // sum over each 16-lane half
}

// ---------------------------------------------------------------------------
// C[M,N] = A[M,K] @ W[K,N] + bias[N]
// block = 128 threads (4 waves). Macro tile 64M x 64N, Kc = 16, double-
// buffered LDS staging. Wave w computes rows [16w,16w+16) of the tile.
//   As[buf][m][k]  (stride 20: b128 stores 16B aligned, frag b64 8B aligned)
//   Ws[buf][n][k]  (transposed so B-frags are contiguous k-pairs)
// ---------------------------------------------------------------------------
__global__ __launch_bounds__(128) void gemm_bias_f32(
    const float* __restrict__ A, const float* __restrict__ W,
    const float* __restrict__ bias, float* __restrict__ C,
    int M, int N, int K) {
  const int tid  = threadIdx.x;
  const int lane = tid & 31;
  const int wave = tid >> 5;
  const int hi   = lane >> 4;
  const int ln   = lane & 15;
  const int mb   = blockIdx.x * 64;
  const int nb   = blockIdx.y * 64;

  __shared__ float As[2][64][20];
  __shared__ float Ws[2][64][20];

  const v8f zero8 = {0.f, 0.f, 0.f, 0.f, 0.f, 0.f, 0.f, 0.f};
  v8f acc[4];
#pragma unroll
  for (int t = 0; t < 4; ++t) acc[t] = zero8;

  // Cooperative tile load indices (2 float4 per thread for each of A, W).
  const int a_r0 = min(mb + (tid >> 2), M - 1);          // A rows (idx 0..127)
  const int a_r1 = min(mb + ((tid + 128) >> 2), M - 1);  // A rows (idx 128..)
  const int a_c  = (tid & 3) * 4;                        // k within tile
  const int w_k0 = tid >> 4;                             // W row 0..7
  const int w_n  = (tid & 15) * 4;                       // n within tile

  float4 ra0, ra1, rw0, rw1;

#define LOAD_TILES(kb)                                                        \
  do {                                                                        \
    ra0 = *(const float4*)(A + (size_t)a_r0 * K + (kb) + a_c);                \
    ra1 = *(const float4*)(A + (size_t)a_r1 * K + (kb) + a_c);                \
    rw0 = *(const float4*)(W + (size_t)((kb) + w_k0) * N + nb + w_n);         \
    rw1 = *(const float4*)(W + (size_t)((kb) + w_k0 + 8) * N + nb + w_n);     \
  } while (0)

#define STORE_TILES(buf)                                                      \
  do {                                                                        \
    *(float4*)&As[buf][tid >> 2][a_c] = ra0;                                  \
    *(float4*)&As[buf][(tid + 128) >> 2][a_c] = ra1;                          \
    Ws[buf][w_n + 0][w_k0] = rw0.x;                                           \
    Ws[buf][w_n + 1][w_k0] = rw0.y;                                           \
    Ws[buf][w_n + 2][w_k0] = rw0.z;                                           \
    Ws[buf][w_n + 3][w_k0] = rw0.w;                                           \
    Ws[buf][w_n + 0][w_k0 + 8] = rw1.x;                                       \
    Ws[buf][w_n + 1][w_k0 + 8] = rw1.y;                                       \
    Ws[buf][w_n + 2][w_k0 + 8] = rw1.z;                                       \
    Ws[buf][w_n + 3][w_k0 + 8] = rw1.w;                                       \
  } while (0)

  LOAD_TILES(0);
  STORE_TILES(0);
  __syncthreads();

  const int NB = K >> 4;  // Kc = 16
  for (int i = 0; i < NB; ++i) {
    const int cur = i & 1;
    if (i + 1 < NB) LOAD_TILES((i + 1) * 16);  // overlap with compute below

#pragma unroll
    for (int k0 = 0; k0 < 16; k0 += 4) {
      v2f a = *(const v2f*)&As[cur][16 * wave + ln][k0 + 2 * hi];
#pragma unroll
      for (int t = 0; t < 4; ++t) {
        v2f b = *(const v2f*)&Ws[cur][16 * t + ln][k0 + 2 * hi];
        acc[t] = wmma_f32(a, b, acc[t]);
      }
    }

    if (i + 1 < NB) {
      STORE_TILES(cur ^ 1);  // s_wait_loadcnt lands here, after the WMMAs
      __syncthreads();
    }
  }
#undef LOAD_TILES
#undef STORE_TILES

#pragma unroll
  for (int t = 0; t < 4; ++t) {
#pragma unroll
    for (int r = 0; r < 8; ++r) {
      const int m = mb + 16 * wave + r + 8 * hi;  // C/D layout: m = r + 8*hi
      const int n = nb + 16 * t + ln;             // n = lane%16
      if (m < M) C[(size_t)m * N + n] = acc[t][r] + bias[n];
    }
  }
}

// ---------------------------------------------------------------------------
// Flash attention per (b*n, h): O[64,64] = softmax(Q Kh^T / 8) Vh
// Q/K/V/O stored as [rows, 512] with head h at columns [h*64, h*64+64).
// One block = 4 waves; wave w owns query rows [16w, 16w+16).
// K/V chunks (16 keys) staged once per block in double-buffered LDS:
//   Ks[buf][s][e] (stride 68), Vs[buf][e][s] transposed (stride 18).
// ---------------------------------------------------------------------------
__global__ __launch_bounds__(128) void attn_f32(
    const float* __restrict__ Q, const float* __restrict__ Kp,
    const float* __restrict__ Vp, float* __restrict__ O, int S) {
  const int bnh  = blockIdx.x;  // 0..447
  const int h    = bnh & 7;
  const int bn   = bnh >> 3;
  const int tid  = threadIdx.x;
  const int lane = tid & 31;
  const int wave = tid >> 5;
  const int hi   = lane >> 4;
  const int ln   = lane & 15;
  const int l0   = wave * 16;

  __shared__ float Ks[2][16][68];
  __shared__ float Vs[2][64][18];
  __shared__ float pshm[4][16][18];  // wave-private P tile (stride 18: 8B ok)

  // Q fragments in registers, scale 1/sqrt(64) folded in.
  v2f qf[16];
  {
    const v2f* Qv =
        (const v2f*)(Q + (size_t)(bn * 64 + l0 + ln) * 512 + h * 64);
#pragma unroll
    for (int kk = 0; kk < 16; ++kk) qf[kk] = Qv[2 * kk + hi] * 0.125f;
  }

  const v8f zero8 = {0.f, 0.f, 0.f, 0.f, 0.f, 0.f, 0.f, 0.f};
  v8f oacc[4];
#pragma unroll
  for (int t = 0; t < 4; ++t) oacc[t] = zero8;
  float run_max[8], run_sum[8];
#pragma unroll
  for (int r = 0; r < 8; ++r) { run_max[r] = -1e30f; run_sum[r] = 0.f; }

  // Cooperative K/V chunk staging: 16 rows x 64 cols, 2 float4/thread each.
  const int kv_r0 = tid >> 4;        // row 0..7
  const int kv_e  = (tid & 15) * 4;  // e within head
  float4 rk0, rk1, rv0, rv1;

#define LOAD_KV(s0)                                                           \
  do {                                                                        \
    const int sr0 = min((s0) + kv_r0, S - 1);                                 \
    const int sr1 = min((s0) + kv_r0 + 8, S - 1);                             \
    rk0 = *(const float4*)(Kp + (size_t)sr0 * 512 + h * 64 + kv_e);           \
    rk1 = *(const float4*)(Kp + (size_t)sr1 * 512 + h * 64 + kv_e);           \
    rv0 = *(const float4*)(Vp + (size_t)sr0 * 512 + h * 64 + kv_e);           \
    rv1 = *(const float4*)(Vp + (size_t)sr1 * 512 + h * 64 + kv_e);           \
  } while (0)

#define STORE_KV(buf)                                                         \
  do {                                                                        \
    *(float4*)&Ks[buf][kv_r0][kv_e] = rk0;                                    \
    *(float4*)&Ks[buf][kv_r0 + 8][kv_e] = rk1;                                \
    Vs[buf][kv_e + 0][kv_r0] = rv0.x;                                         \
    Vs[buf][kv_e + 1][kv_r0] = rv0.y;                                         \
    Vs[buf][kv_e + 2][kv_r0] = rv0.z;                                         \
    Vs[buf][kv_e + 3][kv_r0] = rv0.w;                                         \
    Vs[buf][kv_e + 0][kv_r0 + 8] = rv1.x;                                     \
    Vs[buf][kv_e + 1][kv_r0 + 8] = rv1.y;                                     \
    Vs[buf][kv_e + 2][kv_r0 + 8] = rv1.z;                                     \
    Vs[buf][kv_e + 3][kv_r0 + 8] = rv1.w;                                     \
  } while (0)

  LOAD_KV(0);
  STORE_KV(0);
  __syncthreads();

  const int nchunks = (S + 15) / 16;  // 63 for S=1000
  for (int c = 0; c < nchunks; ++c) {
    const int s0  = c * 16;
    const int cur = c & 1;
    if (c + 1 < nchunks) LOAD_KV((c + 1) * 16);  // overlap with compute

    // ---- scores tile: D[m=query, n=key] = Q(16x64) @ Kh^T(64x16) ----
    v8f sc = zero8;
#pragma unroll
    for (int kk = 0; kk < 16; ++kk) {
      v2f b = *(const v2f*)&Ks[cur][ln][4 * kk + 2 * hi];
      sc = wmma_f32(qf[kk], b, sc);
    }

    // ---- online softmax (rows r + 8*hi live across lanes n of each half) --
    const bool valid = (s0 + ln) < S;
    float factor[8];
#pragma unroll
    for (int r = 0; r < 8; ++r) {
      float x  = valid ? sc[r] : -1e30f;
      float mx = half_max(x);
      float nm = fmaxf(run_max[r], mx);
      float cr = __expf(run_max[r] - nm);  // 0 on first chunk
      float p  = __expf(x - nm);
      run_sum[r] = run_sum[r] * cr + half_sum(p);
      run_max[r] = nm;
      factor[r]  = cr;
      sc[r] = p;
    }
#pragma unroll
    for (int t = 0; t < 4; ++t)
#pragma unroll
      for (int r = 0; r < 8; ++r) oacc[t][r] *= factor[r];

    // ---- relayout P (D-layout -> A-layout) through wave-private LDS ----
#pragma unroll
    for (int r = 0; r < 8; ++r) pshm[wave][r + 8 * hi][ln] = sc[r];
    // same-wave DS store->load ordered via DScnt by the compiler

    // ---- oacc += P(16x16) @ Vh(16x64) ----
#pragma unroll
    for (int kk = 0; kk < 4; ++kk) {
      v2f a = *(const v2f*)&pshm[wave][ln][4 * kk + 2 * hi];
#pragma unroll
      for (int t = 0; t < 4; ++t) {
        v2f b = *(const v2f*)&Vs[cur][16 * t + ln][4 * kk + 2 * hi];
        oacc[t] = wmma_f32(a, b, oacc[t]);
      }
    }

    if (c + 1 < nchunks) {
      STORE_KV(cur ^ 1);  // global-load wait lands here, after the WMMAs
      __syncthreads();
    }
  }
#undef LOAD_KV
#undef STORE_KV

  // ---- normalize and store ----
#pragma unroll
  for (int r = 0; r < 8; ++r) {
    const float inv = 1.0f / run_sum[r];
    const int l = l0 + r + 8 * hi;
    float* Orow = O + (size_t)(bn * 64 + l) * 512 + h * 64;
#pragma unroll
    for (int t = 0; t < 4; ++t) Orow[16 * t + ln] = oacc[t][r] * inv;
  }
}

// ---------------------------------------------------------------------------
extern "C" void kernel_launch(void* const* d_in, const int* in_sizes, int n_in,
                              void* d_out, int out_size, void* d_ws,
                              size_t ws_size, hipStream_t stream) {
  const float* target = (const float*)d_in[0];   // [8,7,64,512] -> [3584,512]
  const float* source = (const float*)d_in[1];   // [1000,768]
  const float* value  = (const float*)d_in[2];   // [1000,768]
  const float* Wq = (const float*)d_in[3];
  const float* bq = (const float*)d_in[4];
  const float* Wk = (const float*)d_in[5];
  const float* bk = (const float*)d_in[6];
  const float* Wv = (const float*)d_in[7];
  const float* bv = (const float*)d_in[8];
  const float* Wo = (const float*)d_in[9];
  const float* bo = (const float*)d_in[10];
  float* out = (float*)d_out;

  const int M  = 8 * 7 * 64;  // 3584 query tokens
  const int S  = 1000;
  const int E  = 512;         // embed dim
  const int DL = 768;         // d_llm

  char* ws = (char*)d_ws;
  float* qbuf = (float*)ws; ws += (size_t)M * E * sizeof(float);
  float* kbuf = (float*)ws; ws += (size_t)S * E * sizeof(float);
  float* vbuf = (float*)ws; ws += (size_t)S * E * sizeof(float);
  float* obuf = (float*)ws;                       // [3584,512]

  const dim3 blk(128);
  // Projections
  gemm_bias_f32<<<dim3((M + 63) / 64, E / 64), blk, 0, stream>>>(
      target, Wq, bq, qbuf, M, E, E);
  gemm_bias_f32<<<dim3((S + 63) / 64, E / 64), blk, 0, stream>>>(
      source, Wk, bk, kbuf, S, E, DL);
  gemm_bias_f32<<<dim3((S + 63) / 64, E / 64), blk, 0, stream>>>(
      value, Wv, bv, vbuf, S, E, DL);
  // Attention: one block per (b*n, head)
  attn_f32<<<dim3(8 * 7 * 8), blk, 0, stream>>>(qbuf, kbuf, vbuf, obuf, S);
  // Output projection -> d_out [3584, 768]
  gemm_bias_f32<<<dim3((M + 63) / 64, DL / 64), blk, 0, stream>>>(
      obuf, Wo, bo, out, M, DL, E);
}